// LaplacianAttention1D_11742440588160
// MI455X (gfx1250) — compile-verified
//
#include <hip/hip_runtime.h>
#include <math.h>

#define Bb    2
#define Hh    8
#define Nn    2048
#define Mm    2048
#define Dd    64
#define Cc    (Hh*Dd)
#define INV_LAMBDA 0.25f
#define EPSV  1e-6f

typedef float v2f __attribute__((ext_vector_type(2)));
typedef float v8f __attribute__((ext_vector_type(8)));
typedef float v4f __attribute__((ext_vector_type(4)));

// ---------------------------------------------------------------------------
// K1: G[bh,n,m] = exp(-L1(q[n],k[m]) / 4).  64x64 tile per block, 4x4 per thread.
// ---------------------------------------------------------------------------
__global__ __launch_bounds__(256) void k1_gmat(const float* __restrict__ q,
                                               const float* __restrict__ kk,
                                               float* __restrict__ G) {
    __shared__ float Qs[64][65];
    __shared__ float Ks[64][65];
    const int bh = blockIdx.z;
    const int n0 = blockIdx.y * 64;
    const int m0 = blockIdx.x * 64;
    const int t  = threadIdx.x;

    const float* qb = q  + ((size_t)bh * Nn + n0) * Dd;
    const float* kb = kk + ((size_t)bh * Nn + m0) * Dd;
    #pragma unroll
    for (int i = 0; i < 16; ++i) {
        int idx = t + i * 256;
        int row = idx >> 6, col = idx & 63;
        Qs[row][col] = qb[row * Dd + col];
        Ks[row][col] = kb[row * Dd + col];
    }
    __syncthreads();

    const int tx = t & 15, ty = t >> 4;
    float acc[4][4];
    #pragma unroll
    for (int i = 0; i < 4; ++i)
        #pragma unroll
        for (int j = 0; j < 4; ++j) acc[i][j] = 0.f;

    #pragma unroll 4
    for (int d = 0; d < 64; ++d) {
        float qv[4], kv[4];
        #pragma unroll
        for (int i = 0; i < 4; ++i) { qv[i] = Qs[4*ty + i][d]; kv[i] = Ks[4*tx + i][d]; }
        #pragma unroll
        for (int i = 0; i < 4; ++i)
            #pragma unroll
            for (int j = 0; j < 4; ++j)
                acc[i][j] += fabsf(qv[i] - kv[j]);
    }

    float* Gb = G + ((size_t)bh * Nn + n0) * Mm + m0;
    #pragma unroll
    for (int i = 0; i < 4; ++i) {
        v4f o;
        #pragma unroll
        for (int j = 0; j < 4; ++j) o[j] = __expf(acc[i][j] * (-INV_LAMBDA));
        *(v4f*)(Gb + (size_t)(4*ty + i) * Mm + 4*tx) = o;
    }
}

// ---------------------------------------------------------------------------
// K2: rowmean r[row] = (1/M) * sum_m G[row,m].   One wave32 per row.
// ---------------------------------------------------------------------------
__global__ __launch_bounds__(256) void k2_rowmean(const float* __restrict__ G,
                                                  float* __restrict__ r) {
    const int lane = threadIdx.x & 31;
    const int row  = blockIdx.x * 8 + (threadIdx.x >> 5);
    const float* g = G + (size_t)row * Mm;
    float sum = 0.f;
    #pragma unroll 8
    for (int i = lane; i < Mm; i += 32) sum += g[i];
    #pragma unroll
    for (int o = 16; o > 0; o >>= 1) sum += __shfl_xor(sum, o, 32);
    if (lane == 0) r[row] = sum * (1.0f / Mm);
}

// ---------------------------------------------------------------------------
// K3: per-column biased variance over N of (G - r), invstd s[col].
// ---------------------------------------------------------------------------
__global__ __launch_bounds__(256) void k3_colstats(const float* __restrict__ G,
                                                   const float* __restrict__ r,
                                                   float* __restrict__ s) {
    const int col = blockIdx.x * 256 + threadIdx.x;   // 0 .. B*H*M-1
    const int bh  = col >> 11;                        // /Mm
    const int m   = col & (Mm - 1);
    const float* gb = G + (size_t)bh * Nn * Mm + m;
    const float* rb = r + bh * Nn;
    float s1 = 0.f, s2 = 0.f;
    for (int n = 0; n < Nn; ++n) {
        float x = gb[(size_t)n * Mm] - rb[n];
        s1 += x; s2 += x * x;
    }
    float mu  = s1 * (1.0f / Nn);
    float var = s2 * (1.0f / Nn) - mu * mu;
    s[col] = rsqrtf(var + EPSV);
}

// ---------------------------------------------------------------------------
// K4: y = Z @ V (+ depthwise conv1d + bias).  Z normalized on the fly while
// staging into LDS; GEMM via V_WMMA_F32_16X16X4_F32.  Block = 8 waves covering
// a 32(n) x 64(d) output tile; wave (wrow,wcol) owns a 16x16 tile.
// ---------------------------------------------------------------------------
__global__ __launch_bounds__(256) void k4_gemm(const float* __restrict__ G,
                                               const float* __restrict__ r,
                                               const float* __restrict__ s,
                                               const float* __restrict__ v,
                                               const float* __restrict__ w,
                                               const float* __restrict__ bias,
                                               float* __restrict__ y) {
    __shared__ float Zs[32][65];
    __shared__ float Vs[64][65];

    const int bh   = blockIdx.y;
    const int h    = bh & (Hh - 1);
    const int n0   = blockIdx.x * 32;
    const int t    = threadIdx.x;
    const int lane = t & 31;
    const int wv   = t >> 5;
    const int wrow = wv >> 2;       // 0..1
    const int wcol = wv & 3;        // 0..3
    const float invM = 1.0f / Mm;

    const float* rb = r + bh * Nn + n0;
    const float* sb = s + bh * Mm;
    const size_t gbase = ((size_t)bh * Nn + n0) * Mm;
    const size_t vbase = (size_t)bh * Nn * Dd;

    v8f acc = {0.f, 0.f, 0.f, 0.f, 0.f, 0.f, 0.f, 0.f};

    // A fragment addressing (16x4 f32): lanes 0-15 -> K={0,1}, lanes 16-31 -> K={2,3}
    const int arow = 16 * wrow + (lane & 15);
    const int koff = (lane >> 4) * 2;
    // B fragment addressing (4x16 f32): vgpr0 rows {0,1}, vgpr1 rows {2,3}
    const int bcol  = 16 * wcol + (lane & 15);
    const int bkoff = lane >> 4;

    for (int kc = 0; kc < Mm; kc += 64) {
        // stage normalized Z chunk: 32 rows x 64 k
        #pragma unroll
        for (int i = 0; i < 8; ++i) {
            int idx = t + i * 256;
            int row = idx >> 6, k = idx & 63;
            float g = G[gbase + (size_t)row * Mm + kc + k];
            Zs[row][k] = (g - rb[row]) * sb[kc + k] + invM;
        }
        // stage V chunk: 64 k x 64 d
        #pragma unroll
        for (int i = 0; i < 16; ++i) {
            int idx = t + i * 256;
            int krow = idx >> 6, d = idx & 63;
            Vs[krow][d] = v[vbase + (size_t)(kc + krow) * Dd + d];
        }
        if (kc + 64 < Mm)
            __builtin_prefetch(G + gbase + (size_t)(t >> 3) * Mm + kc + 64, 0, 0);
        __syncthreads();

        #pragma unroll
        for (int k4 = 0; k4 < 64; k4 += 4) {
            v2f a, b;
            a.x = Zs[arow][k4 + koff];
            a.y = Zs[arow][k4 + koff + 1];
            b.x = Vs[k4 + bkoff][bcol];
            b.y = Vs[k4 + bkoff + 2][bcol];
            acc = __builtin_amdgcn_wmma_f32_16x16x4_f32(
                      false, a, false, b, (short)0, acc, false, false);
        }
        __syncthreads();
    }

    // Epilogue: depthwise conv1d over N (kernel 3, zero pad) + bias, fused store.
    const int dcol = 16 * wcol + (lane & 15);
    const int c    = h * Dd + dcol;
    const float w0 = w[c * 3 + 0], w1 = w[c * 3 + 1], w2 = w[c * 3 + 2];
    const float bi = bias[c];
    #pragma unroll
    for (int i = 0; i < 8; ++i) {
        int n = n0 + 16 * wrow + i + 8 * (lane >> 4);
        float conv = bi + w1 * v[vbase + (size_t)n * Dd + dcol];
        if (n > 0)      conv += w0 * v[vbase + (size_t)(n - 1) * Dd + dcol];
        if (n < Nn - 1) conv += w2 * v[vbase + (size_t)(n + 1) * Dd + dcol];
        y[((size_t)bh * Nn + n) * Dd + dcol] = acc[i] + conv;
    }
}

// ---------------------------------------------------------------------------
extern "C" void kernel_launch(void* const* d_in, const int* in_sizes, int n_in,
                              void* d_out, int out_size, void* d_ws, size_t ws_size,
                              hipStream_t stream) {
    const float* q    = (const float*)d_in[0];
    const float* k    = (const float*)d_in[1];
    const float* v    = (const float*)d_in[2];
    const float* dw_w = (const float*)d_in[3];  // [C,1,3]
    const float* dw_b = (const float*)d_in[4];  // [C]
    float* y = (float*)d_out;

    // workspace layout: r (B*H*N f32), s (B*H*M f32), G (B*H*N*M f32)
    float* r = (float*)d_ws;
    float* s = r + (size_t)Bb * Hh * Nn;
    float* G = (float*)((char*)d_ws + 256 * 1024);   // 256KB-aligned offset past r,s

    dim3 g1(Mm / 64, Nn / 64, Bb * Hh);
    k1_gmat<<<g1, 256, 0, stream>>>(q, k, G);

    k2_rowmean<<<(Bb * Hh * Nn) / 8, 256, 0, stream>>>(G, r);

    k3_colstats<<<(Bb * Hh * Mm) / 256, 256, 0, stream>>>(G, r, s);

    dim3 g4(Nn / 32, Bb * Hh);
    k4_gemm<<<g4, 256, 0, stream>>>(G, r, s, v, dw_w, dw_b, y);
}